// CrossContext_8899172238143
// MI455X (gfx1250) — compile-verified
//
#include <hip/hip_runtime.h>
#include <hip/hip_bf16.h>
#include <math.h>

// ---------------------------------------------------------------------------
// Shapes (fixed by reference setup_inputs): B=4, C=64, N=2048, K=16
// ---------------------------------------------------------------------------
static constexpr int BB = 4;
static constexpr int CC = 64;
static constexpr int NN = 2048;
static constexpr int KK = 16;
static constexpr int D3 = 3 * CC;      // 192, flattened vector dim for kNN
static constexpr int NCOL = 3 * NN;    // 6144, columns of each projection GEMM

typedef float v2f __attribute__((ext_vector_type(2)));
typedef float v8f __attribute__((ext_vector_type(8)));

// Workspace layout (bytes)
static constexpr size_t OFF_DIST = 0;                                  // 2048*2048*4 (reused per batch)
static constexpr size_t OFF_SQ   = OFF_DIST + (size_t)NN * NN * 4;     // 8192*4
static constexpr size_t OFF_IDX  = OFF_SQ + (size_t)BB * NN * 4;       // 131072*4
static constexpr size_t OFF_W    = OFF_IDX + (size_t)BB * NN * KK * 4; // 10*4096*4
static constexpr size_t OFF_PROJ = OFF_W + (size_t)10 * CC * CC * 4;   // 10*B*64*3*2048*4
static constexpr size_t OFF_STAT = OFF_PROJ + (size_t)10 * BB * CC * 3 * NN * 4; // 384*4
static constexpr size_t OFF_YT   = OFF_STAT + 2048;                    // B*N*192*4 (transposed y)

// proj element (a, b, c, v, n):  proj[ ((a*4+b)*64 + c)*3*2048 + v*2048 + n ]
__device__ __forceinline__ const float* parr(const float* proj, int a, int b, int c) {
  return proj + ((((size_t)a * BB + b) * CC + c) * 3) * (size_t)NN;
}

// ---------------------------------------------------------------------------
// 1. Effective weights: Wq, Dq, Wk1, (Wk2-Wk1), Dk1, (Dk2-Dk1), Wv1, (Wv2-Wv1),
//    Dv1, (Dv2-Dv1)  -- each [64][64]
// ---------------------------------------------------------------------------
__global__ __launch_bounds__(256) void prep_weights_kernel(
    const float* __restrict__ Wq, const float* __restrict__ Dq,
    const float* __restrict__ Wk, const float* __restrict__ Dk,
    const float* __restrict__ Wv, const float* __restrict__ Dv,
    float* __restrict__ weff) {
  int t = blockIdx.x * 256 + threadIdx.x;          // 0 .. 40959
  if (t >= 10 * CC * CC) return;
  int m = t >> 12;                                 // matrix 0..9
  int e = t & 4095;
  int o = e >> 6, c = e & 63;
  float v;
  switch (m) {
    case 0: v = Wq[o * 64 + c]; break;
    case 1: v = Dq[o * 64 + c]; break;
    case 2: v = Wk[o * 128 + c]; break;
    case 3: v = Wk[o * 128 + 64 + c] - Wk[o * 128 + c]; break;
    case 4: v = Dk[o * 128 + c]; break;
    case 5: v = Dk[o * 128 + 64 + c] - Dk[o * 128 + c]; break;
    case 6: v = Wv[o * 128 + c]; break;
    case 7: v = Wv[o * 128 + 64 + c] - Wv[o * 128 + c]; break;
    case 8: v = Dv[o * 128 + c]; break;
    default: v = Dv[o * 128 + 64 + c] - Dv[o * 128 + c]; break;
  }
  weff[t] = v;
}

// ---------------------------------------------------------------------------
// 2a. Transpose y -> yT[b][n][192] so WMMA fragments load as aligned b64.
// ---------------------------------------------------------------------------
__global__ __launch_bounds__(256) void transpose_kernel(const float* __restrict__ y,
                                                        float* __restrict__ yT) {
  __shared__ float tile[32][33];
  const int b = blockIdx.z;
  const int d0 = blockIdx.y * 32;
  const int n0 = blockIdx.x * 32;
  const int tx = threadIdx.x;        // 0..31
  const int ty = threadIdx.y;        // 0..7
  const float* src = y + (size_t)b * D3 * NN;
  float* dst = yT + (size_t)b * NN * D3;
#pragma unroll
  for (int i = ty; i < 32; i += 8)
    tile[i][tx] = src[(size_t)(d0 + i) * NN + n0 + tx];
  __syncthreads();
#pragma unroll
  for (int i = ty; i < 32; i += 8)
    dst[(size_t)(n0 + i) * D3 + d0 + tx] = tile[tx][i];
}

// ---------------------------------------------------------------------------
// 2b. sq[b][n] = sum over 192 dims of y^2 (for kNN ranking)
// ---------------------------------------------------------------------------
__global__ __launch_bounds__(256) void sqnorm_kernel(const float* __restrict__ y,
                                                     float* __restrict__ sq) {
  int t = blockIdx.x * 256 + threadIdx.x;
  if (t >= BB * NN) return;
  int b = t >> 11, n = t & (NN - 1);
  const float* p = y + (size_t)b * D3 * NN + n;
  float s = 0.f;
  for (int d = 0; d < D3; ++d) {
    float v = p[(size_t)d * NN];
    s += v * v;
  }
  sq[t] = s;
}

// ---------------------------------------------------------------------------
// 3a. Distance tiles via V_WMMA_F32_16X16X4_F32 on the transposed layout.
//     One wave computes a 16(row) x 64(col) strip: A-frag loaded once per
//     K-step (b64), reused across 4 column tiles (4 b64 B-frags, 4 WMMAs).
//     Output value: 2*inner(n,m) - sq[m]  (row-constant -sq[n] dropped;
//     does not affect per-row top-k ranking).
// ---------------------------------------------------------------------------
__global__ __launch_bounds__(32) void dist_kernel(const float* __restrict__ yT,
                                                  const float* __restrict__ sq,
                                                  float* __restrict__ dist, int b) {
  const int lane = threadIdx.x;
  const int half = lane >> 4;
  const int l = lane & 15;
  const int m0 = blockIdx.x * 64;   // column strip
  const int n0 = blockIdx.y * 16;   // row tile
  const float* base = yT + (size_t)b * NN * D3;

  const float* arow = base + (size_t)(n0 + l) * D3;
  const float* br0 = base + (size_t)(m0 + l) * D3;
  const float* br1 = br0 + 16 * D3;
  const float* br2 = br0 + 32 * D3;
  const float* br3 = br0 + 48 * D3;

  v8f acc0 = {}, acc1 = {}, acc2 = {}, acc3 = {};
  for (int d0 = 0; d0 < D3; d0 += 4) {
    const int o = d0 + 2 * half;   // lanes 0-15: K0,K1 ; lanes 16-31: K2,K3
    v2f a = *(const v2f*)(arow + o);
    v2f b0 = *(const v2f*)(br0 + o);
    v2f b1 = *(const v2f*)(br1 + o);
    v2f b2 = *(const v2f*)(br2 + o);
    v2f b3 = *(const v2f*)(br3 + o);
    acc0 = __builtin_amdgcn_wmma_f32_16x16x4_f32(false, a, false, b0, (short)0, acc0, false, false);
    acc1 = __builtin_amdgcn_wmma_f32_16x16x4_f32(false, a, false, b1, (short)0, acc1, false, false);
    acc2 = __builtin_amdgcn_wmma_f32_16x16x4_f32(false, a, false, b2, (short)0, acc2, false, false);
    acc3 = __builtin_amdgcn_wmma_f32_16x16x4_f32(false, a, false, b3, (short)0, acc3, false, false);
  }

  float sq0 = sq[b * NN + m0 + l];
  float sq1 = sq[b * NN + m0 + 16 + l];
  float sq2 = sq[b * NN + m0 + 32 + l];
  float sq3 = sq[b * NN + m0 + 48 + l];
  float* drow = dist + (size_t)(n0 + half * 8) * NN + m0 + l;
#pragma unroll
  for (int r = 0; r < 8; ++r) {
    drow[(size_t)r * NN + 0]  = 2.f * acc0[r] - sq0;
    drow[(size_t)r * NN + 16] = 2.f * acc1[r] - sq1;
    drow[(size_t)r * NN + 32] = 2.f * acc2[r] - sq2;
    drow[(size_t)r * NN + 48] = 2.f * acc3[r] - sq3;
  }
}

// ---------------------------------------------------------------------------
// 3b. Top-16 per row. One wave per row (4 waves / block). Per-lane sorted
//     top-16 over a strided scan, then lane 0 does a 32-way merge.
// ---------------------------------------------------------------------------
__global__ __launch_bounds__(128) void topk_kernel(const float* __restrict__ dist,
                                                   int* __restrict__ idx, int b) {
  __shared__ float svals[4][32][16];
  __shared__ int sidx[4][32][16];
  __shared__ int scur[4][32];
  const int wave = threadIdx.x >> 5;
  const int lane = threadIdx.x & 31;
  const int row = blockIdx.x * 4 + wave;
  const float* dr = dist + (size_t)row * NN;

  float bv[16];
  int bi[16];
#pragma unroll
  for (int i = 0; i < 16; ++i) { bv[i] = -3.4e38f; bi[i] = 0x7fffffff; }

  for (int j = lane; j < NN; j += 32) {
    float v = dr[j];
    if (v > bv[15] || (v == bv[15] && j < bi[15])) {
      bv[15] = v; bi[15] = j;
#pragma unroll
      for (int i = 15; i >= 1; --i) {
        bool sw = (bv[i] > bv[i - 1]) || (bv[i] == bv[i - 1] && bi[i] < bi[i - 1]);
        if (sw) {
          float tv = bv[i]; bv[i] = bv[i - 1]; bv[i - 1] = tv;
          int ti = bi[i]; bi[i] = bi[i - 1]; bi[i - 1] = ti;
        }
      }
    }
  }
#pragma unroll
  for (int i = 0; i < 16; ++i) { svals[wave][lane][i] = bv[i]; sidx[wave][lane][i] = bi[i]; }
  scur[wave][lane] = 0;
  __syncthreads();

  if (lane == 0) {
    int* outp = idx + ((size_t)b * NN + row) * KK;
    for (int t = 0; t < 16; ++t) {
      float best = -3.4e38f; int bl = 0, bj = 0x7fffffff;
      for (int l2 = 0; l2 < 32; ++l2) {
        int cu = scur[wave][l2];
        if (cu < 16) {
          float v = svals[wave][l2][cu];
          int jj = sidx[wave][l2][cu];
          if (v > best || (v == best && jj < bj)) { best = v; bl = l2; bj = jj; }
        }
      }
      outp[t] = bj;
      scur[wave][bl]++;
    }
  }
}

// ---------------------------------------------------------------------------
// 4. Projection GEMMs: Out[64][6144] = W[64][64] * X[64][6144] per (mat,batch).
//    One wave per column tile computes all 4 row tiles (M-blocking): the
//    strided B-fragment is loaded once per K-step and reused by 4 WMMAs;
//    weight A-fragments are contiguous b64 loads from a cache-hot 16 KB matrix.
//    grid = (384 col-tiles, 40 = mat*4+b).
// ---------------------------------------------------------------------------
__global__ __launch_bounds__(32) void gemm64_kernel(const float* __restrict__ x,
                                                    const float* __restrict__ y,
                                                    const float* __restrict__ weff,
                                                    float* __restrict__ proj) {
  const int lane = threadIdx.x;
  const int half = lane >> 4;
  const int l = lane & 15;
  const int col0 = blockIdx.x * 16;
  const int mb = blockIdx.y;
  const int mat = mb >> 2, b = mb & 3;

  const float* X = (mat < 2 ? x : y) + (size_t)b * CC * 3 * NN;  // [64][6144]
  const float* W = weff + mat * CC * CC;
  float* O = proj + (((size_t)mat * BB + b) * CC) * 3 * (size_t)NN;

  v8f acc[4] = {{}, {}, {}, {}};
  for (int c0 = 0; c0 < CC; c0 += 4) {
    const float* xc = X + (size_t)(c0 + 2 * half) * NCOL + col0 + l;
    v2f bfr;
    bfr.x = xc[0];
    bfr.y = xc[NCOL];
#pragma unroll
    for (int t = 0; t < 4; ++t) {
      v2f a = *(const v2f*)(W + (size_t)(16 * t + l) * CC + c0 + 2 * half);
      acc[t] = __builtin_amdgcn_wmma_f32_16x16x4_f32(false, a, false, bfr, (short)0,
                                                     acc[t], false, false);
    }
  }
#pragma unroll
  for (int t = 0; t < 4; ++t) {
    float* orow = O + (size_t)(16 * t + half * 8) * NCOL + col0 + l;
#pragma unroll
    for (int r = 0; r < 8; ++r) orow[(size_t)r * NCOL] = acc[t][r];
  }
}

// ---------------------------------------------------------------------------
// 5. BatchNorm statistics (sum, sumsq of ||p||+EPS per output channel)
// ---------------------------------------------------------------------------
__device__ __forceinline__ float reduceBlock256(float v, float* red) {
  int t = threadIdx.x;
  red[t] = v;
  __syncthreads();
  for (int s = 128; s > 0; s >>= 1) {
    if (t < s) red[t] += red[t + s];
    __syncthreads();
  }
  float r = red[0];
  __syncthreads();
  return r;
}

__global__ __launch_bounds__(256) void stats_q_kernel(const float* __restrict__ proj,
                                                      float* __restrict__ stat) {
  __shared__ float red[256];
  int t = blockIdx.x * 256 + threadIdx.x;        // B*64*N = 524288
  int n = t & (NN - 1);
  int c = (t >> 11) & 63;
  int b = t >> 17;
  const float* P = parr(proj, 0, b, c);
  float vx = P[n], vy = P[NN + n], vz = P[2 * NN + n];
  float nr = sqrtf(vx * vx + vy * vy + vz * vz) + 1e-6f;
  float s1 = reduceBlock256(nr, red);
  float s2 = reduceBlock256(nr * nr, red);
  if (threadIdx.x == 0) {
    atomicAdd(&stat[2 * c], s1);
    atomicAdd(&stat[2 * c + 1], s2);
  }
}

__global__ __launch_bounds__(256) void stats_kv_kernel(const float* __restrict__ proj,
                                                       const int* __restrict__ idx,
                                                       float* __restrict__ stat) {
  __shared__ float red[256];
  int t = blockIdx.x * 256 + threadIdx.x;        // B*64*N*K = 8388608
  int kk = t & 15;
  int n = (t >> 4) & (NN - 1);
  int c = (t >> 15) & 63;
  int b = t >> 21;
  int j = idx[((size_t)b * NN + n) * KK + kk];

  const float* K1 = parr(proj, 2, b, c);
  const float* K2 = parr(proj, 3, b, c);
  float kx = K1[j] + K2[n], ky = K1[NN + j] + K2[NN + n], kz = K1[2 * NN + j] + K2[2 * NN + n];
  float nK = sqrtf(kx * kx + ky * ky + kz * kz) + 1e-6f;

  const float* V1 = parr(proj, 6, b, c);
  const float* V2 = parr(proj, 7, b, c);
  float vx = V1[j] + V2[n], vy = V1[NN + j] + V2[NN + n], vz = V1[2 * NN + j] + V2[2 * NN + n];
  float nV = sqrtf(vx * vx + vy * vy + vz * vz) + 1e-6f;

  float s1 = reduceBlock256(nK, red);
  float s2 = reduceBlock256(nK * nK, red);
  float s3 = reduceBlock256(nV, red);
  float s4 = reduceBlock256(nV * nV, red);
  if (threadIdx.x == 0) {
    atomicAdd(&stat[128 + 2 * c], s1);
    atomicAdd(&stat[128 + 2 * c + 1], s2);
    atomicAdd(&stat[256 + 2 * c], s3);
    atomicAdd(&stat[256 + 2 * c + 1], s4);
  }
}

// ---------------------------------------------------------------------------
// 6. Fused final kernel: one block (64 threads = 64 channels) per (b,n).
// ---------------------------------------------------------------------------
__device__ __forceinline__ void bn_leaky(float3& p, const float3 d, float mu, float rs,
                                         float g, float be) {
  float nr = sqrtf(p.x * p.x + p.y * p.y + p.z * p.z) + 1e-6f;
  float s = ((nr - mu) * rs * g + be) / nr;
  p.x *= s; p.y *= s; p.z *= s;
  float dot = p.x * d.x + p.y * d.y + p.z * d.z;
  float dns = d.x * d.x + d.y * d.y + d.z * d.z + 1e-6f;
  float f = (dot < 0.f) ? (0.8f * dot / dns) : 0.f;  // (1-NEG) = 0.8
  p.x -= f * d.x; p.y -= f * d.y; p.z -= f * d.z;
}

__device__ __forceinline__ float reduce64(float* red, int c, float v) {
  red[c] = v;
  __syncthreads();
  for (int s = 32; s > 0; s >>= 1) {
    if (c < s) red[c] += red[c + s];
    __syncthreads();
  }
  float r = red[0];
  __syncthreads();
  return r;
}

__global__ __launch_bounds__(64) void final_kernel(
    const float* __restrict__ x, const float* __restrict__ proj,
    const int* __restrict__ idx, const float* __restrict__ stat,
    const float* __restrict__ gq, const float* __restrict__ bq,
    const float* __restrict__ gk, const float* __restrict__ bk,
    const float* __restrict__ gv, const float* __restrict__ bv,
    float* __restrict__ out) {
  __shared__ float red[64];
  const int bn = blockIdx.x;
  const int b = bn >> 11;
  const int n = bn & (NN - 1);
  const int c = threadIdx.x;

  const float icQ = 1.f / (BB * (float)NN);
  const float icKV = 1.f / (BB * (float)NN * KK);
  float muQ = stat[2 * c] * icQ;
  float rsQ = rsqrtf(stat[2 * c + 1] * icQ - muQ * muQ + 1e-5f);
  float muK = stat[128 + 2 * c] * icKV;
  float rsK = rsqrtf(stat[128 + 2 * c + 1] * icKV - muK * muK + 1e-5f);
  float muV = stat[256 + 2 * c] * icKV;
  float rsV = rsqrtf(stat[256 + 2 * c + 1] * icKV - muV * muV + 1e-5f);
  float gqc = gq[c], bqc = bq[c], gkc = gk[c], bkc = bk[c], gvc = gv[c], bvc = bv[c];

  // ---- Q path: BN + leaky + channel-equivariant normalize
  const float* Pq = parr(proj, 0, b, c);
  const float* Dq = parr(proj, 1, b, c);
  float3 p = make_float3(Pq[n], Pq[NN + n], Pq[2 * NN + n]);
  float3 d = make_float3(Dq[n], Dq[NN + n], Dq[2 * NN + n]);
  bn_leaky(p, d, muQ, rsQ, gqc, bqc);
  float tot = reduce64(red, c, p.x * p.x + p.y * p.y + p.z * p.z);
  float inv = 1.f / fmaxf(sqrtf(tot), 1e-12f);
  float3 q = make_float3(p.x * inv, p.y * inv, p.z * inv);

  // ---- center-dependent parts (fixed for this n)
  const float* K1 = parr(proj, 2, b, c);
  const float* K2 = parr(proj, 3, b, c);
  const float* DK1 = parr(proj, 4, b, c);
  const float* DK2 = parr(proj, 5, b, c);
  const float* V1 = parr(proj, 6, b, c);
  const float* V2 = parr(proj, 7, b, c);
  const float* DV1 = parr(proj, 8, b, c);
  const float* DV2 = parr(proj, 9, b, c);
  float3 k2 = make_float3(K2[n], K2[NN + n], K2[2 * NN + n]);
  float3 dk2 = make_float3(DK2[n], DK2[NN + n], DK2[2 * NN + n]);
  float3 v2 = make_float3(V2[n], V2[NN + n], V2[2 * NN + n]);
  float3 dv2 = make_float3(DV2[n], DV2[NN + n], DV2[2 * NN + n]);
  const int* id = idx + ((size_t)b * NN + n) * KK;

  float qk[KK], vyx[KK], vyy[KK], vyz[KK];
#pragma unroll
  for (int kk = 0; kk < KK; ++kk) {
    int j = id[kk];
    float3 pk = make_float3(K1[j] + k2.x, K1[NN + j] + k2.y, K1[2 * NN + j] + k2.z);
    float3 dk = make_float3(DK1[j] + dk2.x, DK1[NN + j] + dk2.y, DK1[2 * NN + j] + dk2.z);
    bn_leaky(pk, dk, muK, rsK, gkc, bkc);
    float t2 = reduce64(red, c, pk.x * pk.x + pk.y * pk.y + pk.z * pk.z);
    float ik = 1.f / fmaxf(sqrtf(t2), 1e-12f);
    qk[kk] = (pk.x * q.x + pk.y * q.y + pk.z * q.z) * ik;

    float3 pv = make_float3(V1[j] + v2.x, V1[NN + j] + v2.y, V1[2 * NN + j] + v2.z);
    float3 dv = make_float3(DV1[j] + dv2.x, DV1[NN + j] + dv2.y, DV1[2 * NN + j] + dv2.z);
    bn_leaky(pv, dv, muV, rsV, gvc, bvc);
    vyx[kk] = pv.x; vyy[kk] = pv.y; vyz[kk] = pv.z;
  }

  // ---- softmax over neighbors (scale 1/sqrt(3*C) = 1/sqrt(192))
  const float scale = 0.07216878364870323f;
  float m = -3.4e38f;
#pragma unroll
  for (int kk = 0; kk < KK; ++kk) m = fmaxf(m, qk[kk] * scale);
  float s = 0.f;
#pragma unroll
  for (int kk = 0; kk < KK; ++kk) { qk[kk] = expf(qk[kk] * scale - m); s += qk[kk]; }
  float is = 1.f / s;
  float ax = 0.f, ay = 0.f, az = 0.f;
#pragma unroll
  for (int kk = 0; kk < KK; ++kk) {
    float a = qk[kk] * is;
    ax += a * vyx[kk]; ay += a * vyy[kk]; az += a * vyz[kk];
  }

  size_t ob = (((size_t)b * CC + c) * 3) * (size_t)NN + n;
  out[ob] = x[ob] + ax;
  out[ob + NN] = x[ob + NN] + ay;
  out[ob + 2 * NN] = x[ob + 2 * NN] + az;
}

// ---------------------------------------------------------------------------
// Launch
// ---------------------------------------------------------------------------
extern "C" void kernel_launch(void* const* d_in, const int* in_sizes, int n_in,
                              void* d_out, int out_size, void* d_ws, size_t ws_size,
                              hipStream_t stream) {
  (void)in_sizes; (void)n_in; (void)out_size; (void)ws_size;
  const float* x = (const float*)d_in[0];
  const float* y = (const float*)d_in[1];
  const float* Wq = (const float*)d_in[2];
  const float* Dq = (const float*)d_in[3];
  const float* gq = (const float*)d_in[4];
  const float* bq = (const float*)d_in[5];
  const float* Wk = (const float*)d_in[6];
  const float* Dk = (const float*)d_in[7];
  const float* gk = (const float*)d_in[8];
  const float* bk = (const float*)d_in[9];
  const float* Wv = (const float*)d_in[10];
  const float* Dv = (const float*)d_in[11];
  const float* gv = (const float*)d_in[12];
  const float* bv = (const float*)d_in[13];
  float* out = (float*)d_out;

  char* ws = (char*)d_ws;
  float* dist = (float*)(ws + OFF_DIST);
  float* sq = (float*)(ws + OFF_SQ);
  int* idx = (int*)(ws + OFF_IDX);
  float* weff = (float*)(ws + OFF_W);
  float* proj = (float*)(ws + OFF_PROJ);
  float* stat = (float*)(ws + OFF_STAT);
  float* yT = (float*)(ws + OFF_YT);

  // stats accumulators must be zero at every launch
  hipMemsetAsync(stat, 0, 3 * CC * 2 * sizeof(float), stream);

  prep_weights_kernel<<<(10 * CC * CC + 255) / 256, 256, 0, stream>>>(Wq, Dq, Wk, Dk, Wv,
                                                                      Dv, weff);
  transpose_kernel<<<dim3(NN / 32, D3 / 32, BB), dim3(32, 8), 0, stream>>>(y, yT);
  sqnorm_kernel<<<(BB * NN + 255) / 256, 256, 0, stream>>>(y, sq);

  // kNN per batch, reusing the dist buffer (stream-ordered)
  for (int b = 0; b < BB; ++b) {
    dist_kernel<<<dim3(NN / 64, NN / 16), 32, 0, stream>>>(yT, sq, dist, b);
    topk_kernel<<<NN / 4, 128, 0, stream>>>(dist, idx, b);
  }

  // 10 projection GEMMs x 4 batches (all 4 M-tiles per wave)
  gemm64_kernel<<<dim3(NCOL / 16, 10 * BB), 32, 0, stream>>>(x, y, weff, proj);

  stats_q_kernel<<<(BB * CC * NN) / 256, 256, 0, stream>>>(proj, stat);
  stats_kv_kernel<<<(BB * CC * NN * KK) / 256, 256, 0, stream>>>(proj, idx, stat);

  final_kernel<<<BB * NN, CC, 0, stream>>>(x, proj, idx, stat, gq, bq, gk, bk, gv, bv,
                                           out);
}